// gauss_net_10685878633070
// MI455X (gfx1250) — compile-verified
//
#include <hip/hip_runtime.h>
#include <cstdint>
#include <cstddef>

// Problem constants (from the reference): B=16, H=W=299, K=8
constexpr int    KNN     = 8;
constexpr int    HW      = 299 * 299;                 // 89401 pixels per image
constexpr int    NB      = 16;
constexpr long long HWK  = (long long)HW * KNN;       // 715208  (weights->indices plane offset)
constexpr long long BSTR = 2 * HWK;                   // 1430416 (per-batch stride in weight_and_index_list)
constexpr size_t NPIX    = (size_t)NB * HW;           // 1430416

// Output tuple layout (flattened, reference return order)
constexpr size_t X_OFF    = 0;                        // x        (B,H,W,4)
constexpr size_t XR_OFF   = NPIX * 4;                 // x_rgba   (B,H,W,4)
constexpr size_t CLA_OFF  = XR_OFF + NPIX * 4;        // cla      (B,3,H,W)
constexpr size_t ORI_OFF  = CLA_OFF + (size_t)NB * 3 * HW;  // ori_img copy (B,H,W,4)
constexpr size_t OCLA_OFF = ORI_OFF + NPIX * 4;       // ori_cla  (B,3,H,W)

typedef __attribute__((address_space(3))) float4 lds_float4;

__global__ __launch_bounds__(256)
void gauss_net_kernel(const float* __restrict__ srgb,   // (200000,4)
                      const float* __restrict__ wi,     // (B,2,H,W,K)
                      const float* __restrict__ ori,    // (B,H,W,4)
                      float* __restrict__ out)
{
    // Per-lane staging tiles: 8 weights + 8 indices per pixel (16 KB LDS / block)
    __shared__ float4 wtile[512];
    __shared__ float4 itile[512];

    const int t  = threadIdx.x;
    const int b  = blockIdx.y;
    const int r  = blockIdx.x * 256 + t;
    const int rc = (r < HW) ? r : (HW - 1);     // clamp so staging stays in-bounds, EXEC stays uniform

    // --- Async stage weight + index streams into LDS (CDNA5 async-to-LDS path) ---
    const uint64_t sbase = (uint64_t)(uintptr_t)wi;
    const uint32_t woff  = (uint32_t)(((uint64_t)b * (uint64_t)BSTR + (uint64_t)rc * KNN) * 4ull);
    const uint32_t ioff  = woff + (uint32_t)(HWK * 4);
    const uint32_t wlds  = (uint32_t)(uintptr_t)(lds_float4*)(&wtile[2 * t]);
    const uint32_t ilds  = (uint32_t)(uintptr_t)(lds_float4*)(&itile[2 * t]);

    asm volatile("global_load_async_to_lds_b128 %0, %1, %2"
                 :: "v"(wlds),        "v"(woff),        "s"(sbase) : "memory");
    asm volatile("global_load_async_to_lds_b128 %0, %1, %2"
                 :: "v"(wlds + 16u),  "v"(woff + 16u),  "s"(sbase) : "memory");
    asm volatile("global_load_async_to_lds_b128 %0, %1, %2"
                 :: "v"(ilds),        "v"(ioff),        "s"(sbase) : "memory");
    asm volatile("global_load_async_to_lds_b128 %0, %1, %2"
                 :: "v"(ilds + 16u),  "v"(ioff + 16u),  "s"(sbase) : "memory");
    // Each lane reads back only the bytes it staged itself -> per-wave counter wait suffices.
    asm volatile("s_wait_asynccnt 0x0" ::: "memory");

    if (r >= HW) return;

    const float4 wv0 = wtile[2 * t + 0];
    const float4 wv1 = wtile[2 * t + 1];
    const float4 iv0 = itile[2 * t + 0];
    const float4 iv1 = itile[2 * t + 1];

    const float wgt[KNN] = {wv0.x, wv0.y, wv0.z, wv0.w, wv1.x, wv1.y, wv1.z, wv1.w};
    const float idf[KNN] = {iv0.x, iv0.y, iv0.z, iv0.w, iv1.x, iv1.y, iv1.z, iv1.w};

    // 8 random float4 gathers; table is 3.2 MB -> L2-resident on the 192 MB L2.
    const float4* __restrict__ s4 = (const float4*)srgb;
    float4 g[KNN];
#pragma unroll
    for (int k = 0; k < KNN; ++k) g[k] = s4[(int)idf[k]];

    float ax = 0.f, ay = 0.f, az = 0.f, aw = 0.f;
#pragma unroll
    for (int k = 0; k < KNN; ++k) {
        ax = fmaf(wgt[k], g[k].x, ax);
        ay = fmaf(wgt[k], g[k].y, ay);
        az = fmaf(wgt[k], g[k].z, az);
        aw = fmaf(wgt[k], g[k].w, aw);
    }

    const size_t pix = (size_t)b * HW + (size_t)r;
    const float4 o = ((const float4*)ori)[pix];

    const float alpha = aw * (1.0f / 255.0f);
    float xr = fmaf(ax, alpha, o.x);
    float xg = fmaf(ay, alpha, o.y);
    float xb = fmaf(az, alpha, o.z);

    const bool vis = (o.w > 0.0f);
    xr = vis ? xr : 0.0f;
    xg = vis ? xg : 0.0f;
    xb = vis ? xb : 0.0f;

    const float cr = fminf(fmaxf(xr,  0.0f), 255.0f);
    const float cg = fminf(fmaxf(xg,  0.0f), 255.0f);
    const float cb = fminf(fmaxf(xb,  0.0f), 255.0f);
    const float ca = fminf(fmaxf(o.w, 0.0f), 255.0f);

    // x (unclipped weighted sum)
    ((float4*)(out + X_OFF))[pix]  = make_float4(ax, ay, az, aw);
    // x_rgba
    ((float4*)(out + XR_OFF))[pix] = make_float4(cr, cg, cb, ca);
    // ori_img passthrough
    ((float4*)(out + ORI_OFF))[pix] = o;

    // NCHW alpha-to-white planes (coalesced along W)
    float* __restrict__ cla  = out + CLA_OFF  + (size_t)b * 3 * HW;
    float* __restrict__ ocla = out + OCLA_OFF + (size_t)b * 3 * HW;
    cla [(size_t)0 * HW + r] = vis ? cr  : 255.0f;
    cla [(size_t)1 * HW + r] = vis ? cg  : 255.0f;
    cla [(size_t)2 * HW + r] = vis ? cb  : 255.0f;
    ocla[(size_t)0 * HW + r] = vis ? o.x : 255.0f;
    ocla[(size_t)1 * HW + r] = vis ? o.y : 255.0f;
    ocla[(size_t)2 * HW + r] = vis ? o.z : 255.0f;
}

extern "C" void kernel_launch(void* const* d_in, const int* in_sizes, int n_in,
                              void* d_out, int out_size, void* d_ws, size_t ws_size,
                              hipStream_t stream) {
    (void)in_sizes; (void)n_in; (void)out_size; (void)d_ws; (void)ws_size;
    const float* srgb = (const float*)d_in[0];  // spatial_rgb
    const float* wi   = (const float*)d_in[1];  // weight_and_index_list
    const float* ori  = (const float*)d_in[2];  // ori_img
    float* out        = (float*)d_out;

    dim3 grid((HW + 255) / 256, NB);
    gauss_net_kernel<<<grid, 256, 0, stream>>>(srgb, wi, ori, out);
}